// HungarianMatcherDynamicK_33071248179974
// MI455X (gfx1250) — compile-verified
//
#include <hip/hip_runtime.h>
#include <hip/hip_bf16.h>

#define BQ 64
#define QN 2000
#define GN 128
#define CN 80
#define BIGF 3.0e38f

typedef float v2f __attribute__((ext_vector_type(2)));
typedef float v8f __attribute__((ext_vector_type(8)));

// ---------------------------------------------------------------------------
// Kernel 1: fused cost matrix.  cost_class via V_WMMA_F32_16X16X4_F32:
//   cost_class = PN[Q x C] @ OneHot(labels)[C x G]   (bit-exact gather-GEMM)
// Block = 256 threads (8 waves); tile = 16 queries x 128 gts (all of G, so the
// per-row fg mask can be reduced inside the block).
// ---------------------------------------------------------------------------
__global__ __launch_bounds__(256) void k_cost(
    const float* __restrict__ logits, const float* __restrict__ pboxes,
    const float* __restrict__ gboxes, const int* __restrict__ glabels,
    const float* __restrict__ imgsz, const float* __restrict__ imgszt,
    float* __restrict__ cost)
{
  const int b   = blockIdx.y;
  const int q0  = blockIdx.x * 16;
  const int tid = threadIdx.x;
  const int wid = tid >> 5;
  const int lane = tid & 31;
  const int hi  = lane >> 4;       // 0: lanes 0-15, 1: lanes 16-31
  const int lm  = lane & 15;

  __shared__ float pn[16][84];     // padded stride (84*4 bytes, conflict-light)
  __shared__ float qb[16][4];
  __shared__ float nqb[16][4];
  __shared__ float gb[GN][4];
  __shared__ float ngb[GN][4];
  __shared__ int   glab[GN];
  __shared__ unsigned int fg_bits; // bit m set => row m is foreground

  if (tid == 0) fg_bits = 0u;

  // focal pos-neg table for this 16-query tile
  for (int idx = tid; idx < 16 * CN; idx += 256) {
    int r = idx / CN, c = idx % CN;
    float x = logits[((size_t)b * QN + q0 + r) * CN + c];
    float p = 1.0f / (1.0f + __expf(-x));
    float pos = 0.25f * (1.0f - p) * (1.0f - p) * (-__logf(p + 1e-8f));
    float neg = 0.75f * p * p * (-__logf(1.0f - p + 1e-8f));
    pn[r][c] = pos - neg;
  }
  for (int idx = tid; idx < 16 * 4; idx += 256) {
    int r = idx >> 2, c = idx & 3;
    float v = pboxes[((size_t)b * QN + q0 + r) * 4 + c];
    qb[r][c] = v;
    nqb[r][c] = v / imgsz[b * 4 + c];
  }
  for (int idx = tid; idx < GN * 4; idx += 256) {
    int r = idx >> 2, c = idx & 3;
    float v = gboxes[((size_t)b * GN + r) * 4 + c];
    gb[r][c] = v;
    ngb[r][c] = v / imgszt[((size_t)b * GN + r) * 4 + c];
  }
  for (int idx = tid; idx < GN; idx += 256) glab[idx] = glabels[b * GN + idx];
  __syncthreads();

  // ---- WMMA: 16x16 tile of cost_class, K = C = 80 (20 steps of k=4) ----
  const int g0  = wid * 16;
  const int myg = g0 + lm;          // this lane's column
  const int mylab = glab[myg];

  v8f acc = {0.f, 0.f, 0.f, 0.f, 0.f, 0.f, 0.f, 0.f};
  for (int k0 = 0; k0 < CN; k0 += 4) {
    const int ka = k0 + 2 * hi;     // K = k0 + j + 2*hi (j = vector component)
    v2f a, bb;
    a.x = pn[lm][ka];
    a.y = pn[lm][ka + 1];
    bb.x = (mylab == ka)     ? 1.0f : 0.0f;
    bb.y = (mylab == ka + 1) ? 1.0f : 0.0f;
    acc = __builtin_amdgcn_wmma_f32_16x16x4_f32(
        false, a, false, bb, (short)0, acc, false, false);
  }

  // ---- per-pair geometric costs on top of the WMMA accumulator ----
  const float gx0 = gb[myg][0], gy0 = gb[myg][1], gx1 = gb[myg][2], gy1 = gb[myg][3];
  const float ng0 = ngb[myg][0], ng1 = ngb[myg][1], ng2 = ngb[myg][2], ng3 = ngb[myg][3];
  const float gcx = 0.5f * (gx0 + gx1), gcy = 0.5f * (gy0 + gy1);
  const float rw = 2.5f * (gx1 - gx0), rh = 2.5f * (gy1 - gy0);
  const float areaB = (gx1 - gx0) * (gy1 - gy0);

  float cst[8];
  unsigned int pam = 0u;            // (in_box | in_center) bits per tile row
  for (int v = 0; v < 8; ++v) {
    const int mr = v + 8 * hi;      // D layout: M = v + 8*hi, N = lm
    float ax0 = qb[mr][0], ay0 = qb[mr][1], ax1 = qb[mr][2], ay1 = qb[mr][3];
    float ax = 0.5f * (ax0 + ax1), ay = 0.5f * (ay0 + ay1);
    float areaA = (ax1 - ax0) * (ay1 - ay0);
    float iw = fmaxf(fminf(ax1, gx1) - fmaxf(ax0, gx0), 0.f);
    float ih = fmaxf(fminf(ay1, gy1) - fmaxf(ay0, gy0), 0.f);
    float inter = iw * ih;
    float uni = areaA + areaB - inter;
    float iou = inter / uni;
    float cw = fmaxf(fmaxf(ax1, gx1) - fminf(ax0, gx0), 0.f);
    float ch = fmaxf(fmaxf(ay1, gy1) - fminf(ay0, gy0), 0.f);
    float carea = cw * ch;
    float giou = iou - (carea - uni) / carea;
    float cb = fabsf(nqb[mr][0] - ng0) + fabsf(nqb[mr][1] - ng1) +
               fabsf(nqb[mr][2] - ng2) + fabsf(nqb[mr][3] - ng3);
    bool inb = (ax > gx0) && (ax < gx1) && (ay > gy0) && (ay < gy1);
    bool inc = (ax > gcx - rw) && (ax < gcx + rw) && (ay > gcy - rh) && (ay < gcy + rh);
    if (inb || inc) pam |= (1u << mr);
    cst[v] = 5.0f * cb + 2.0f * acc[v] - 2.0f * giou + ((inb && inc) ? 0.f : 100.f);
  }

  // fg = OR over all 128 gts, per query row: OR within each 16-lane half,
  // then combine the 8 waves through LDS.
  for (int o = 1; o < 16; o <<= 1) pam |= __shfl_xor(pam, o, 32);
  if (lm == 0) atomicOr(&fg_bits, pam);
  __syncthreads();
  const unsigned int fb = fg_bits;

  for (int v = 0; v < 8; ++v) {
    const int mr = v + 8 * hi;
    float c = cst[v] + (((fb >> mr) & 1u) ? 0.f : 10000.f);
    cost[((size_t)b * QN + q0 + mr) * GN + myg] = c;
  }
}

// ---------------------------------------------------------------------------
// Kernel 2: dyn_k[g] = max(1, floor(sum of top-5 IoU over Q)).  One wave per
// (b, g); lane-strided top-5, merged through LDS.
// ---------------------------------------------------------------------------
__global__ __launch_bounds__(128) void k_dynk(
    const float* __restrict__ pboxes, const float* __restrict__ gboxes,
    int* __restrict__ dynks)
{
  const int b = blockIdx.y;
  const int wid = threadIdx.x >> 5, lane = threadIdx.x & 31;
  const int g = blockIdx.x * 4 + wid;
  __shared__ float t5[4][160];

  const float gx0 = gboxes[((size_t)b * GN + g) * 4 + 0];
  const float gy0 = gboxes[((size_t)b * GN + g) * 4 + 1];
  const float gx1 = gboxes[((size_t)b * GN + g) * 4 + 2];
  const float gy1 = gboxes[((size_t)b * GN + g) * 4 + 3];
  const float areaB = (gx1 - gx0) * (gy1 - gy0);

  float best[5] = {0.f, 0.f, 0.f, 0.f, 0.f};   // ious are >= 0
  for (int q = lane; q < QN; q += 32) {
    const float4 a = reinterpret_cast<const float4*>(pboxes)[(size_t)b * QN + q];
    float areaA = (a.z - a.x) * (a.w - a.y);
    float iw = fmaxf(fminf(a.z, gx1) - fmaxf(a.x, gx0), 0.f);
    float ih = fmaxf(fminf(a.w, gy1) - fmaxf(a.y, gy0), 0.f);
    float inter = iw * ih;
    float iou = inter / (areaA + areaB - inter);
    if (iou > best[4]) {
      best[4] = iou;
      for (int i = 4; i > 0; --i)
        if (best[i] > best[i - 1]) { float t = best[i - 1]; best[i - 1] = best[i]; best[i] = t; }
    }
  }
  for (int j = 0; j < 5; ++j) t5[wid][lane * 5 + j] = best[j];
  __syncthreads();
  if (lane == 0) {
    float s = 0.f;
    for (int t = 0; t < 5; ++t) {
      float bv = -1.f; int bj = 0;
      for (int j = 0; j < 160; ++j)
        if (t5[wid][j] > bv) { bv = t5[wid][j]; bj = j; }
      s += bv;
      t5[wid][bj] = -1.f;
    }
    int k = (int)s;          // truncation toward zero, s in [0,5]
    if (k < 1) k = 1;
    dynks[b * GN + g] = k;
  }
}

// ---------------------------------------------------------------------------
// Kernel 3: per-batch serial dynamic-K assignment + repair loop.
// m kept as a 32KB row-major LDS bitset; cost stays L2-resident.
// Penalty is row-constant, so row argmin (fix target) is loop-invariant.
// ---------------------------------------------------------------------------
__device__ inline void lexmin32(float& bv, int& bq) {
  for (int o = 16; o > 0; o >>= 1) {
    float ov = __shfl_xor(bv, o, 32);
    int   oq = __shfl_xor(bq, o, 32);
    if (ov < bv || (ov == bv && oq < bq)) { bv = ov; bq = oq; }
  }
}

__global__ __launch_bounds__(256) void k_assign(
    const float* __restrict__ cost, const int* __restrict__ dynks,
    float* __restrict__ out_sel, float* __restrict__ out_gti,
    float* __restrict__ out_mq, float* __restrict__ out_m)
{
  const int b = blockIdx.x;
  const int tid = threadIdx.x, wid = tid >> 5, lane = tid & 31;
  const float* C = cost + (size_t)b * QN * GN;

  __shared__ unsigned int mrow[QN][4];           // 32 KB bitset: m[q][g]
  __shared__ int row_sum[QN];
  __shared__ unsigned short pen[QN];             // #iterations row was matched
  __shared__ unsigned char rarg[QN];             // row argmin column (invariant)
  __shared__ unsigned int staleb[(QN + 31) / 32];
  __shared__ int col_sum[GN];
  __shared__ float selc[8][160];
  __shared__ int   selq[8][160];
  __shared__ int f_any_zero, f_do_fix;

  for (int q = tid; q < QN; q += 256) {
    mrow[q][0] = mrow[q][1] = mrow[q][2] = mrow[q][3] = 0u;
    row_sum[q] = 0; pen[q] = 0;
  }
  for (int i = tid; i < GN; i += 256) col_sum[i] = 0;
  for (int i = tid; i < (QN + 31) / 32; i += 256) staleb[i] = 0u;
  __syncthreads();

  // row argmin over columns (first-min, matches jnp.argmin)
  for (int q = tid; q < QN; q += 256) {
    const float* r = C + (size_t)q * GN;
    float bv = r[0]; int bg = 0;
    for (int g = 1; g < GN; ++g) { float v = r[g]; if (v < bv) { bv = v; bg = g; } }
    rarg[q] = (unsigned char)bg;
  }
  __syncthreads();

  // per-column lexicographic top-k assignment (stable tie-break = smaller q)
  for (int i = 0; i < 16; ++i) {
    const int g = wid * 16 + i;
    float bc[5] = {BIGF, BIGF, BIGF, BIGF, BIGF};
    int   bq[5] = {0x7fffffff, 0x7fffffff, 0x7fffffff, 0x7fffffff, 0x7fffffff};
    for (int q = lane; q < QN; q += 32) {
      float v = C[(size_t)q * GN + g];
      if (v < bc[4] || (v == bc[4] && q < bq[4])) {
        bc[4] = v; bq[4] = q;
        for (int j = 4; j > 0; --j)
          if (bc[j] < bc[j - 1] || (bc[j] == bc[j - 1] && bq[j] < bq[j - 1])) {
            float tc = bc[j - 1]; bc[j - 1] = bc[j]; bc[j] = tc;
            int tq = bq[j - 1]; bq[j - 1] = bq[j]; bq[j] = tq;
          }
      }
    }
    for (int j = 0; j < 5; ++j) { selc[wid][lane * 5 + j] = bc[j]; selq[wid][lane * 5 + j] = bq[j]; }
    __syncthreads();
    if (lane == 0) {
      int k = dynks[b * GN + g]; if (k > 5) k = 5;
      for (int t = 0; t < k; ++t) {
        float bv = BIGF; int bqq = 0x7fffffff, bj = 0;
        for (int j = 0; j < 160; ++j) {
          float v = selc[wid][j]; int qq = selq[wid][j];
          if (v < bv || (v == bv && qq < bqq)) { bv = v; bqq = qq; bj = j; }
        }
        selc[wid][bj] = BIGF; selq[wid][bj] = 0x7fffffff;
        atomicOr(&mrow[bqq][g >> 5], 1u << (g & 31));
      }
      col_sum[g] = k;
    }
    __syncthreads();
  }

  // row sums + stale set (from the initial m, before row_fix — as reference)
  for (int q = tid; q < QN; q += 256) {
    int rs = __popc(mrow[q][0]) + __popc(mrow[q][1]) + __popc(mrow[q][2]) + __popc(mrow[q][3]);
    row_sum[q] = rs;
    if (rs > 1) atomicOr(&staleb[q >> 5], 1u << (q & 31));
  }
  __syncthreads();

  // initial row_fix for stale rows
  for (int q = tid; q < QN; q += 256) {
    if ((staleb[q >> 5] >> (q & 31)) & 1u) {
      for (int w = 0; w < 4; ++w) {
        unsigned int bits = mrow[q][w];
        while (bits) { int g = w * 32 + __ffs((int)bits) - 1; bits &= bits - 1; atomicSub(&col_sum[g], 1); }
        mrow[q][w] = 0u;
      }
      int g = rarg[q];
      mrow[q][g >> 5] = 1u << (g & 31);
      atomicAdd(&col_sum[g], 1);
      row_sum[q] = 1;
    }
  }
  __syncthreads();

  // repair loop (<= 50 iterations, exactly the reference body order)
  for (int it = 0; it < 50; ++it) {
    if (tid == 0) { f_any_zero = 0; f_do_fix = 0; }
    __syncthreads();
    for (int g = tid; g < GN; g += 256) if (col_sum[g] == 0) atomicOr(&f_any_zero, 1);
    __syncthreads();
    if (!f_any_zero) break;

    for (int q = tid; q < QN; q += 256) if (row_sum[q] > 0) pen[q]++;
    __syncthreads();

    // assign each zero column its argmin row of (cost + 1e5*pen)
    for (int i = 0; i < 16; ++i) {
      const int g = wid * 16 + i;
      if (col_sum[g] == 0) {
        float bv = BIGF; int bq = 0x7fffffff;
        for (int q = lane; q < QN; q += 32) {
          float v = C[(size_t)q * GN + g] + 100000.0f * (float)pen[q];
          if (v < bv || (v == bv && q < bq)) { bv = v; bq = q; }
        }
        lexmin32(bv, bq);
        if (lane == 0) {
          atomicOr(&mrow[bq][g >> 5], 1u << (g & 31));
          col_sum[g] = 1;
          atomicAdd(&row_sum[bq], 1);
        }
      }
    }
    __syncthreads();
    for (int q = tid; q < QN; q += 256) if (row_sum[q] > 1) atomicOr(&f_do_fix, 1);
    __syncthreads();
    if (f_do_fix) {
      for (int q = tid; q < QN; q += 256) {
        if ((staleb[q >> 5] >> (q & 31)) & 1u) {
          for (int w = 0; w < 4; ++w) {
            unsigned int bits = mrow[q][w];
            while (bits) { int g = w * 32 + __ffs((int)bits) - 1; bits &= bits - 1; atomicSub(&col_sum[g], 1); }
            mrow[q][w] = 0u;
          }
          int g = rarg[q];
          mrow[q][g >> 5] = 1u << (g & 31);
          atomicAdd(&col_sum[g], 1);
          row_sum[q] = 1;
        }
      }
      __syncthreads();
    }
  }
  __syncthreads();

  // outputs
  for (int q = tid; q < QN; q += 256) {
    out_sel[(size_t)b * QN + q] = (row_sum[q] > 0) ? 1.0f : 0.0f;
    int gi = 0;
    for (int w = 0; w < 4; ++w)
      if (mrow[q][w]) { gi = w * 32 + __ffs((int)mrow[q][w]) - 1; break; }
    out_gti[(size_t)b * QN + q] = (float)gi;
  }
  for (int i = 0; i < 16; ++i) {
    const int g = wid * 16 + i;
    float bv = BIGF; int bq = 0x7fffffff;
    for (int q = lane; q < QN; q += 32) {
      if ((mrow[q][g >> 5] >> (g & 31)) & 1u) {
        float v = C[(size_t)q * GN + g] + 100000.0f * (float)pen[q];
        if (v < bv || (v == bv && q < bq)) { bv = v; bq = q; }
      }
    }
    lexmin32(bv, bq);
    if (lane == 0) out_mq[(size_t)b * GN + g] = (bv >= BIGF) ? 0.0f : (float)bq;
  }
  for (int idx = tid; idx < QN * GN; idx += 256) {
    int q = idx >> 7, g = idx & 127;
    out_m[(size_t)b * QN * GN + idx] = ((mrow[q][g >> 5] >> (g & 31)) & 1u) ? 1.0f : 0.0f;
  }
}

// ---------------------------------------------------------------------------
extern "C" void kernel_launch(void* const* d_in, const int* in_sizes, int n_in,
                              void* d_out, int out_size, void* d_ws, size_t ws_size,
                              hipStream_t stream) {
  const float* logits = (const float*)d_in[0];
  const float* pboxes = (const float*)d_in[1];
  const int*   glab   = (const int*)d_in[2];
  const float* gboxes = (const float*)d_in[3];
  const float* imgsz  = (const float*)d_in[4];
  const float* imgszt = (const float*)d_in[5];

  float* out = (float*)d_out;
  float* out_sel = out;
  float* out_gti = out + (size_t)BQ * QN;
  float* out_mq  = out + 2 * (size_t)BQ * QN;
  float* out_m   = out + 2 * (size_t)BQ * QN + (size_t)BQ * GN;

  float* cost = (float*)d_ws;                                   // 65.5 MB
  int* dynks  = (int*)((char*)d_ws + (size_t)BQ * QN * GN * sizeof(float));

  k_cost  <<<dim3(QN / 16, BQ), 256, 0, stream>>>(logits, pboxes, gboxes, glab, imgsz, imgszt, cost);
  k_dynk  <<<dim3(GN / 4, BQ), 128, 0, stream>>>(pboxes, gboxes, dynks);
  k_assign<<<dim3(BQ), 256, 0, stream>>>(cost, dynks, out_sel, out_gti, out_mq, out_m);
}